// RotatedBboxesLoss_54468775248531
// MI455X (gfx1250) — compile-verified
//
#include <hip/hip_runtime.h>
#include <math.h>

// Problem constants from the reference
#define BV      32
#define LV      8400
#define NCV     15
#define NTOT    (BV * LV)        // 268800 = 1050 * 256 exactly
#define EPS_IOU 0.1f
#define DEG2RAD 0.017453292519943295f

typedef float v2f __attribute__((ext_vector_type(2)));
typedef float v8f __attribute__((ext_vector_type(8)));
typedef unsigned long long u64;

// ---------------------------------------------------------------------------
// Exact 32-lane wave sum on the matrix pipe (v_wmma_f32_16x16x4_f32).
// A-layout (16x4 f32): lane L holds row M = L%16; lanes 0-15 carry K={0,1},
// lanes 16-31 carry K={2,3}. With a = {x, 0} and B = all-ones (layout
// irrelevant), D[m][n] = x[m] + x[m+16] for all n. C/D layout: lanes 0-15 hold
// rows 0..7 in regs 0..7, lanes 16-31 hold rows 8..15, so
// sum(c[0..7])@lane0 + sum(c[0..7])@lane16 is the full 32-lane sum.
// ---------------------------------------------------------------------------
__device__ __forceinline__ float wave_sum32(float x) {
    v2f a; a[0] = x;    a[1] = 0.0f;
    v2f b; b[0] = 1.0f; b[1] = 1.0f;
    v8f c = {};
    c = __builtin_amdgcn_wmma_f32_16x16x4_f32(
            /*neg_a=*/false, a, /*neg_b=*/false, b,
            /*c_mod=*/(short)0, c, /*reuse_a=*/false, /*reuse_b=*/false);
    float s = c[0] + c[1] + c[2] + c[3] + c[4] + c[5] + c[6] + c[7];
    return __shfl(s, 0, 32) + __shfl(s, 16, 32);
}

// Order-preserving float -> uint32 map (total order identical to f32 <).
__device__ __forceinline__ unsigned int ordkey(float f) {
    unsigned int u = __float_as_uint(f);
    return (u & 0x80000000u) ? ~u : (u | 0x80000000u);
}

// ---------------------------------------------------------------------------
// Rotated-box IoU, exact polygon clipping, mirrors the JAX reference.
// Fully register-resident: all 24-vertex arrays are accessed only with
// compile-time-constant indices (loops unrolled; fixed sorting network).
// bx = {cx, cy, w, h, angle_rad}
// ---------------------------------------------------------------------------
__device__ float rotated_iou(const float bx1[5], const float bx2[5]) {
    const float xo[4] = { 0.5f, -0.5f, -0.5f,  0.5f};
    const float yo[4] = {-0.5f, -0.5f,  0.5f,  0.5f};
    float c1x[4], c1y[4], c2x[4], c2y[4];
    {
        float s, c;
        sincosf(bx1[4], &s, &c);
        #pragma unroll
        for (int j = 0; j < 4; ++j) {
            float px = bx1[2] * xo[j], py = bx1[3] * yo[j];
            c1x[j] = c * px - s * py + bx1[0];
            c1y[j] = s * px + c * py + bx1[1];
        }
        sincosf(bx2[4], &s, &c);
        #pragma unroll
        for (int j = 0; j < 4; ++j) {
            float px = bx2[2] * xo[j], py = bx2[3] * yo[j];
            c2x[j] = c * px - s * py + bx2[0];
            c2y[j] = s * px + c * py + bx2[1];
        }
    }

    float vx[24], vy[24];
    unsigned int vmask = 0;   // bit k == vertex k valid

    // corners of box1 inside box2 (verts 0..3)
    {
        float ax = c2x[0], ay = c2y[0];
        float abx = c2x[1] - ax, aby = c2y[1] - ay;
        float adx = c2x[3] - ax, ady = c2y[3] - ay;
        float nab = abx * abx + aby * aby, nad = adx * adx + ady * ady;
        #pragma unroll
        for (int j = 0; j < 4; ++j) {
            float apx = c1x[j] - ax, apy = c1y[j] - ay;
            float pab = apx * abx + apy * aby;
            float pad = apx * adx + apy * ady;
            vx[j] = c1x[j]; vy[j] = c1y[j];
            bool in = (pab >= 0.f) && (pab <= nab) && (pad >= 0.f) && (pad <= nad);
            vmask |= ((unsigned)in) << j;
        }
    }
    // corners of box2 inside box1 (verts 4..7)
    {
        float ax = c1x[0], ay = c1y[0];
        float abx = c1x[1] - ax, aby = c1y[1] - ay;
        float adx = c1x[3] - ax, ady = c1y[3] - ay;
        float nab = abx * abx + aby * aby, nad = adx * adx + ady * ady;
        #pragma unroll
        for (int j = 0; j < 4; ++j) {
            float apx = c2x[j] - ax, apy = c2y[j] - ay;
            float pab = apx * abx + apy * aby;
            float pad = apx * adx + apy * ady;
            vx[4 + j] = c2x[j]; vy[4 + j] = c2y[j];
            bool in = (pab >= 0.f) && (pab <= nab) && (pad >= 0.f) && (pad <= nad);
            vmask |= ((unsigned)in) << (4 + j);
        }
    }
    // edge-edge intersections (verts 8..23)
    #pragma unroll
    for (int j = 0; j < 4; ++j) {
        float p1x = c1x[j], p1y = c1y[j];
        float d1x = c1x[(j + 1) & 3] - p1x, d1y = c1y[(j + 1) & 3] - p1y;
        #pragma unroll
        for (int l = 0; l < 4; ++l) {
            float q1x = c2x[l], q1y = c2y[l];
            float d2x = c2x[(l + 1) & 3] - q1x, d2y = c2y[(l + 1) & 3] - q1y;
            float rx = q1x - p1x, ry = q1y - p1y;
            float den = d1x * d2y - d1y * d2x;
            bool  dn  = fabsf(den) > 1e-12f;
            float sf  = dn ? den : 1.0f;
            float t = (rx * d2y - ry * d2x) / sf;
            float u = (rx * d1y - ry * d1x) / sf;
            bool valid = dn && (t > 0.f) && (t < 1.f) && (u > 0.f) && (u < 1.f);
            int k = 8 + j * 4 + l;
            vx[k] = valid ? (p1x + t * d1x) : 0.f;
            vy[k] = valid ? (p1y + t * d1y) : 0.f;
            vmask |= ((unsigned)valid) << k;
        }
    }

    // centroid of valid vertices
    float cnt = 0.f, mx = 0.f, my = 0.f;
    #pragma unroll
    for (int k = 0; k < 24; ++k) {
        float m = ((vmask >> k) & 1u) ? 1.f : 0.f;
        cnt += m; mx += vx[k] * m; my += vy[k] * m;
    }
    float inv = 1.0f / fmaxf(cnt, 1.0f);
    mx *= inv; my *= inv;

    // Composite 64-bit sort key: (ordered(angle) << 8) | original_index.
    // Ordering is exactly a *stable* ascending sort on the f32 angles,
    // including the 1e6 sentinels for invalid vertices (jnp.argsort semantics).
    u64 key[24];
    #pragma unroll
    for (int k = 0; k < 24; ++k) {
        float a = ((vmask >> k) & 1u) ? atan2f(vy[k] - my, vx[k] - mx) : 1.0e6f;
        key[k] = (((u64)ordkey(a)) << 8) | (u64)(unsigned)k;
    }

    // Odd-even transposition sorting network: 24 rounds of fixed-index,
    // branch-free compare-exchanges -> everything stays in VGPRs (no scratch).
    #pragma unroll
    for (int r = 0; r < 24; ++r) {
        #pragma unroll
        for (int i = (r & 1); i < 23; i += 2) {
            bool sw = key[i] > key[i + 1];
            u64   ka = key[i], kb = key[i + 1];
            float xa = vx[i],  xb = vx[i + 1];
            float ya = vy[i],  yb = vy[i + 1];
            key[i]     = sw ? kb : ka;
            key[i + 1] = sw ? ka : kb;
            vx[i]      = sw ? xb : xa;
            vx[i + 1]  = sw ? xa : xb;
            vy[i]      = sw ? yb : ya;
            vy[i + 1]  = sw ? ya : yb;
        }
    }

    // collapse invalid (sorted-to-the-end) vertices onto the first vertex
    const u64 inval = ((u64)ordkey(1.0e6f)) << 8;
    float fx = vx[0], fy = vy[0];
    #pragma unroll
    for (int k = 0; k < 24; ++k) {
        bool bad = key[k] >= inval;
        vx[k] = bad ? fx : vx[k];
        vy[k] = bad ? fy : vy[k];
    }

    // shoelace
    float sh = 0.f;
    #pragma unroll
    for (int k = 0; k < 24; ++k) {
        int kn = (k + 1 < 24) ? (k + 1) : 0;
        sh += vx[k] * vy[kn] - vx[kn] * vy[k];
    }
    float inter = 0.5f * fabsf(sh);
    float a1 = bx1[2] * bx1[3], a2 = bx2[2] * bx2[3];
    return inter / (a1 + a2 - inter + 1e-9f);
}

// ---------------------------------------------------------------------------
__global__ void rbox_zero_kernel(float* acc) {
    acc[0] = 0.0f;
    acc[1] = 0.0f;
}

__global__ void __launch_bounds__(256)
rbox_loss_main_kernel(const float* __restrict__ pred_bboxes,
                      const float* __restrict__ pred_angles,
                      const float* __restrict__ target_bboxes,
                      const float* __restrict__ target_angles,
                      const float* __restrict__ target_scores,
                      const unsigned char* __restrict__ fg_mask,
                      float* __restrict__ acc) {
    int i = blockIdx.x * 256 + threadIdx.x;
    int j = (i < NTOT) ? i : 0;          // keep EXEC all-ones for WMMA
    float gate = (i < NTOT) ? 1.f : 0.f;

    float mfv = fg_mask[j] ? 1.f : 0.f;

    const float* ts = target_scores + (size_t)j * NCV;
    float bw = 0.f;
    #pragma unroll
    for (int c = 0; c < NCV; ++c) bw += ts[c];

    float b1[5], b2[5];
    const float* pb = pred_bboxes + (size_t)j * 4;
    const float* tb = target_bboxes + (size_t)j * 4;
    b1[0] = pb[0]; b1[1] = pb[1]; b1[2] = pb[2]; b1[3] = pb[3];
    b1[4] = pred_angles[j];
    b2[0] = tb[0]; b2[1] = tb[1]; b2[2] = tb[2]; b2[3] = tb[3];
    b2[4] = target_angles[j] * DEG2RAD;

    float iou  = fmaxf(rotated_iou(b1, b2), EPS_IOU);
    float loss = (1.0f - iou) * bw * mfv * gate;
    mfv *= gate;

    // wave32 reduction on the matrix pipe (v_wmma_f32_16x16x4_f32)
    float wl = wave_sum32(loss);
    float wm = wave_sum32(mfv);

    __shared__ float red[2][8];
    int wid = threadIdx.x >> 5, lane = threadIdx.x & 31;
    if (lane == 0) { red[0][wid] = wl; red[1][wid] = wm; }
    __syncthreads();
    if (threadIdx.x == 0) {
        float a = 0.f, b = 0.f;
        #pragma unroll
        for (int w = 0; w < 8; ++w) { a += red[0][w]; b += red[1][w]; }
        atomicAdd(&acc[0], a);   // global_atomic_add_f32
        atomicAdd(&acc[1], b);
    }
}

__global__ void rbox_finalize_kernel(const float* __restrict__ acc,
                                     const float* __restrict__ tss,
                                     float* __restrict__ out) {
    float lsum    = acc[0];
    float num_pos = acc[1];
    float lmean   = lsum / fmaxf(num_pos, 1.0f);
    float li      = (tss[0] == 0.0f) ? lsum : lmean;
    li            = (num_pos > 0.0f) ? li : 0.0f;
    out[0] = li;      // loss_iou
    out[1] = 0.0f;    // loss_dfl == pred_dist.sum() * 0.0
}

// ---------------------------------------------------------------------------
extern "C" void kernel_launch(void* const* d_in, const int* in_sizes, int n_in,
                              void* d_out, int out_size, void* d_ws, size_t ws_size,
                              hipStream_t stream) {
    (void)in_sizes; (void)n_in; (void)out_size; (void)ws_size;

    // setup_inputs() order:
    // 0: pred_dist (unused: *0.0), 1: pred_bboxes, 2: pred_angles,
    // 3: anchor_points (unused),   4: target_bboxes, 5: target_angles,
    // 6: target_scores, 7: target_scores_sum, 8: fg_mask (numpy bool)
    const float* pred_bboxes   = (const float*)d_in[1];
    const float* pred_angles   = (const float*)d_in[2];
    const float* target_bboxes = (const float*)d_in[4];
    const float* target_angles = (const float*)d_in[5];
    const float* target_scores = (const float*)d_in[6];
    const float* tss           = (const float*)d_in[7];
    const unsigned char* fg    = (const unsigned char*)d_in[8];

    float* acc = (float*)d_ws;   // [lsum, num_pos]
    float* out = (float*)d_out;

    rbox_zero_kernel<<<1, 1, 0, stream>>>(acc);
    rbox_loss_main_kernel<<<(NTOT + 255) / 256, 256, 0, stream>>>(
        pred_bboxes, pred_angles, target_bboxes, target_angles,
        target_scores, fg, acc);
    rbox_finalize_kernel<<<1, 1, 0, stream>>>(acc, tss, out);
}